// PointGNN_34222299414580
// MI455X (gfx1250) — compile-verified
//
#include <hip/hip_runtime.h>
#include <hip/hip_bf16.h>

typedef __attribute__((ext_vector_type(2))) float v2f;
typedef __attribute__((ext_vector_type(4))) float v4f;
typedef __attribute__((ext_vector_type(8))) float v8f;

#define DIN  128
#define DOUT 128

// fp32 WMMA: D(16x16) = A(16x4) * B(4x16) + C
__device__ __forceinline__ v8f wmma_f32_k4(v2f a, v2f b, v8f c) {
    return __builtin_amdgcn_wmma_f32_16x16x4_f32(
        /*neg_a=*/false, a, /*neg_b=*/false, b,
        /*c_mod=*/(short)0, c, /*reuse_a=*/false, /*reuse_b=*/false);
}

// CDNA5 no-return float atomic max, immediate byte offset, device scope.
template <int OFF>
__device__ __forceinline__ void atomic_max_f32_off(float* base, float v) {
    asm volatile("global_atomic_max_num_f32 %0, %1, off offset:%c2 scope:SCOPE_DEV"
                 :: "v"(base), "v"(v), "i"(OFF) : "memory");
}

__global__ void pg_init_kernel(float* __restrict__ agg, long total4) {
    long i = (long)blockIdx.x * blockDim.x + threadIdx.x;
    if (i < total4) {
        const float ninf = __int_as_float(0xff800000u);
        v4f v = {ninf, ninf, ninf, ninf};
        ((v4f*)agg)[i] = v;
    }
}

// ---------------------------------------------------------------------------
// Edge kernel: 64 edges per block, 256 threads (8 waves).
//   A[64][128] = vf[src] - vf[dst]   (staged in LDS, b128 granularity)
//   C[64][128] = A @ W_edge + b_edge
//   atomic scatter-max into agg[dst]
// Wave w: M-tile = w>>1 (16 edges), N-half = w&1 (4 x 16 cols).
// ---------------------------------------------------------------------------
__global__ __launch_bounds__(256) void pg_edge_kernel(
    const float* __restrict__ vf, const int* __restrict__ ei,
    const float* __restrict__ We, const float* __restrict__ be,
    float* __restrict__ agg, int N, int E) {
    __shared__ float As[64][132];   // 528B rows: 16B aligned, banks (4m+k)%64
    __shared__ int s_dst[64];
    __shared__ int s_src[64];

    const int tid = threadIdx.x;
    const long e0 = (long)blockIdx.x * 64;

    if (tid < 64) {
        long e = e0 + tid;
        int s = 0, d = 0;
        if (e < E) { s = ei[e]; d = ei[(long)E + e]; }
        s_src[tid] = s; s_dst[tid] = d;
    }
    __syncthreads();

    // gather-subtract into LDS, float4 per thread-iteration (L2-resident vf)
    for (int i = tid; i < 64 * (DIN / 4); i += 256) {   // 2048 chunks, 8 iters
        const int r = i >> 5, c = (i & 31) * 4;
        v4f a = *(const v4f*)&vf[(long)s_src[r] * DIN + c];
        v4f d = *(const v4f*)&vf[(long)s_dst[r] * DIN + c];
        *(v4f*)&As[r][c] = a - d;
    }
    __syncthreads();

    const int w    = tid >> 5;
    const int lane = tid & 31;
    const int lr   = lane & 15;     // row/col within 16
    const int lh   = lane >> 4;     // lane-half
    const int mt   = w >> 1;        // M-tile (16 edges)
    const int nh   = w & 1;         // N half (64 cols)

    v8f acc[4] = {};
    for (int k = 0; k < DIN; k += 4) {
        const int ka = k + 2 * lh;
        // A frag: lane lr holds row M=mt*16+lr, K = ka, ka+1 (8B aligned)
        v2f a = *(const v2f*)&As[mt * 16 + lr][ka];
#pragma unroll
        for (int t = 0; t < 4; ++t) {
            const int n = nh * 64 + t * 16 + lr;
            v2f b;
            b.x = We[(long)ka * DOUT + n];
            b.y = We[(long)(ka + 1) * DOUT + n];
            acc[t] = wmma_f32_k4(a, b, acc[t]);
        }
    }

    // ---- epilogue: bias + scatter-max (C layout: row = r + 8*lh, col = lr)
    const int nbase = nh * 64 + lr;
    const float bias0 = be[nbase];
    const float bias1 = be[nbase + 16];
    const float bias2 = be[nbase + 32];
    const float bias3 = be[nbase + 48];

    int dstm[8];                        // preload this wave's dst rows once
#pragma unroll
    for (int r = 0; r < 8; ++r) dstm[r] = s_dst[mt * 16 + r + 8 * lh];

    if (e0 + 64 <= (long)E) {           // full tile: straight-line atomics
#pragma unroll
        for (int r = 0; r < 8; ++r) {
            float* base = agg + (long)dstm[r] * DOUT + nbase;
            atomic_max_f32_off<0>  (base, acc[0][r] + bias0);
            atomic_max_f32_off<64> (base, acc[1][r] + bias1);
            atomic_max_f32_off<128>(base, acc[2][r] + bias2);
            atomic_max_f32_off<192>(base, acc[3][r] + bias3);
        }
    } else {                            // tail tile: guarded
#pragma unroll
        for (int r = 0; r < 8; ++r) {
            const int m = mt * 16 + r + 8 * lh;
            if (e0 + m < E) {
                float* base = agg + (long)dstm[r] * DOUT + nbase;
                atomic_max_f32_off<0>  (base, acc[0][r] + bias0);
                atomic_max_f32_off<64> (base, acc[1][r] + bias1);
                atomic_max_f32_off<128>(base, acc[2][r] + bias2);
                atomic_max_f32_off<192>(base, acc[3][r] + bias3);
            }
        }
    }
}

// ---------------------------------------------------------------------------
// Node kernel: 32 nodes per block, 256 threads (8 waves).
//   combined[32][256] = [vf | fixup(agg)]
//   h = relu(LN(combined @ W1 + b1))
//   out = h @ W2 + b2
// Wave w: M-tile = w>>2, N-tiles = {2*(w&3), 2*(w&3)+1}
// ---------------------------------------------------------------------------
__global__ __launch_bounds__(256) void pg_node_kernel(
    const float* __restrict__ vf, const float* __restrict__ agg,
    const float* __restrict__ W1, const float* __restrict__ b1,
    const float* __restrict__ gma, const float* __restrict__ bta,
    const float* __restrict__ W2, const float* __restrict__ b2,
    float* __restrict__ out, int N) {
    __shared__ float Cs[32][260];   // 1040B rows: 16B aligned, conflict free
    __shared__ float Hs[32][132];

    const int tid = threadIdx.x;
    const long n0 = (long)blockIdx.x * 32;
    const bool full = (n0 + 32 <= (long)N);

    // stage combined = [vf | agg (-inf -> 0)], float4 chunks (wave-uniform half)
    for (int i = tid; i < 32 * 64; i += 256) {          // 2048 chunks, 8 iters
        const int r = i >> 6, c = (i & 63) * 4;
        long node = n0 + r;
        if (!full && node >= N) node = N - 1;
        v4f v;
        if (c < 128) {
            v = *(const v4f*)&vf[node * DIN + c];
        } else {
            v = *(const v4f*)&agg[node * DOUT + (c - 128)];
#pragma unroll
            for (int j = 0; j < 4; ++j)
                if (!(v[j] >= -1e37f)) v[j] = 0.0f;     // empty segment -> 0
        }
        *(v4f*)&Cs[r][c] = v;
    }
    __syncthreads();

    const int w    = tid >> 5;
    const int lane = tid & 31;
    const int lr   = lane & 15;
    const int lh   = lane >> 4;
    const int mt   = w >> 2;         // 0..1
    const int ntb  = (w & 3) * 2;    // first of two N-tiles

    // GEMM1: [32x256] @ [256x128]
    v8f acc[2] = {};
    for (int k = 0; k < 256; k += 4) {
        const int ka = k + 2 * lh;
        v2f a = *(const v2f*)&Cs[mt * 16 + lr][ka];
#pragma unroll
        for (int t = 0; t < 2; ++t) {
            const int n = (ntb + t) * 16 + lr;
            v2f b;
            b.x = W1[(long)ka * DOUT + n];
            b.y = W1[(long)(ka + 1) * DOUT + n];
            acc[t] = wmma_f32_k4(a, b, acc[t]);
        }
    }
#pragma unroll
    for (int t = 0; t < 2; ++t) {
        const int n = (ntb + t) * 16 + lr;
        const float bias = b1[n];
#pragma unroll
        for (int r = 0; r < 8; ++r)
            Hs[mt * 16 + r + 8 * lh][n] = acc[t][r] + bias;
    }
    __syncthreads();

    // LayerNorm + ReLU: wave w handles rows {w, w+8, w+16, w+24}
    {
        const float g0 = gma[lane],      g1 = gma[lane + 32];
        const float g2 = gma[lane + 64], g3 = gma[lane + 96];
        const float t0 = bta[lane],      t1 = bta[lane + 32];
        const float t2 = bta[lane + 64], t3 = bta[lane + 96];
#pragma unroll
        for (int rr = 0; rr < 4; ++rr) {
            const int m = w + rr * 8;
            float x0 = Hs[m][lane],      x1 = Hs[m][lane + 32];
            float x2 = Hs[m][lane + 64], x3 = Hs[m][lane + 96];
            float s = x0 + x1 + x2 + x3;
            float q = x0 * x0 + x1 * x1 + x2 * x2 + x3 * x3;
#pragma unroll
            for (int off = 16; off > 0; off >>= 1) {
                s += __shfl_xor(s, off, 32);
                q += __shfl_xor(q, off, 32);
            }
            const float mu  = s * (1.0f / 128.0f);
            const float var = q * (1.0f / 128.0f) - mu * mu;
            const float inv = rsqrtf(var + 1e-5f);
            Hs[m][lane]      = fmaxf((x0 - mu) * inv * g0 + t0, 0.0f);
            Hs[m][lane + 32] = fmaxf((x1 - mu) * inv * g1 + t1, 0.0f);
            Hs[m][lane + 64] = fmaxf((x2 - mu) * inv * g2 + t2, 0.0f);
            Hs[m][lane + 96] = fmaxf((x3 - mu) * inv * g3 + t3, 0.0f);
        }
    }
    __syncthreads();

    // GEMM2: [32x128] @ [128x128]
    v8f acc2[2] = {};
    for (int k = 0; k < 128; k += 4) {
        const int ka = k + 2 * lh;
        v2f a = *(const v2f*)&Hs[mt * 16 + lr][ka];
#pragma unroll
        for (int t = 0; t < 2; ++t) {
            const int n = (ntb + t) * 16 + lr;
            v2f b;
            b.x = W2[(long)ka * DOUT + n];
            b.y = W2[(long)(ka + 1) * DOUT + n];
            acc2[t] = wmma_f32_k4(a, b, acc2[t]);
        }
    }

    // store epilogue: uniform fast path for full blocks
    if (full) {
#pragma unroll
        for (int t = 0; t < 2; ++t) {
            const int n = (ntb + t) * 16 + lr;
            const float bias = b2[n];
#pragma unroll
            for (int r = 0; r < 8; ++r) {
                const int m = mt * 16 + r + 8 * lh;
                out[(n0 + m) * DOUT + n] = acc2[t][r] + bias;
            }
        }
    } else {
#pragma unroll
        for (int t = 0; t < 2; ++t) {
            const int n = (ntb + t) * 16 + lr;
            const float bias = b2[n];
#pragma unroll
            for (int r = 0; r < 8; ++r) {
                const int m = mt * 16 + r + 8 * lh;
                const long node = n0 + m;
                if (node < N) out[node * DOUT + n] = acc2[t][r] + bias;
            }
        }
    }
}

extern "C" void kernel_launch(void* const* d_in, const int* in_sizes, int n_in,
                              void* d_out, int out_size, void* d_ws, size_t ws_size,
                              hipStream_t stream) {
    const float* vf  = (const float*)d_in[0];
    const int*   ei  = (const int*)  d_in[1];
    const float* We  = (const float*)d_in[2];
    const float* be  = (const float*)d_in[3];
    const float* W1  = (const float*)d_in[4];
    const float* b1  = (const float*)d_in[5];
    const float* gma = (const float*)d_in[6];
    const float* bta = (const float*)d_in[7];
    const float* W2  = (const float*)d_in[8];
    const float* b2  = (const float*)d_in[9];
    float* out = (float*)d_out;

    const int N = in_sizes[0] / DIN;
    const int E = in_sizes[1] / 2;

    float* agg = (float*)d_ws;              // [N, DOUT] scratch
    const long total4 = (long)N * DOUT / 4;

    pg_init_kernel<<<(int)((total4 + 255) / 256), 256, 0, stream>>>(agg, total4);
    pg_edge_kernel<<<(E + 63) / 64, 256, 0, stream>>>(vf, ei, We, be, agg, N, E);
    pg_node_kernel<<<(N + 31) / 32, 256, 0, stream>>>(vf, agg, W1, b1, gma, bta,
                                                      W2, b2, out, N);
}